// TSIM_14757507629403
// MI455X (gfx1250) — compile-verified
//
#include <hip/hip_runtime.h>
#include <math.h>

#define BB   32
#define CC   56
#define NN   200
#define LL   24
#define NL   (NN * LL)                 // 4800
#define LOUT 12
#define TEN  ((size_t)BB * CC * NN * LL)   // 8,601,600 floats per tensor
#define SMALL ((size_t)BB * CC * LL)       // 43,008 floats

typedef __attribute__((ext_vector_type(2))) float v2f;
typedef __attribute__((ext_vector_type(8))) float v8f;

// ---------------------------------------------------------------------------
// K0: orf[b,c,l] = sum_n outflow[b,c,n,l] - sum_n inflow[b,c,n,l]
// ---------------------------------------------------------------------------
__global__ void orf_kernel(const float* __restrict__ inflow,
                           const float* __restrict__ outflow,
                           float* __restrict__ orf)
{
    int idx = blockIdx.x * blockDim.x + threadIdx.x;
    if (idx >= (int)SMALL) return;
    int l  = idx % LL;
    int bc = idx / LL;
    const float* pin  = inflow  + (size_t)bc * NL + l;
    const float* pout = outflow + (size_t)bc * NL + l;
    float si = 0.f, so = 0.f;
    for (int n = 0; n < NN; ++n) {
        si += pin[(size_t)n * LL];
        so += pout[(size_t)n * LL];
    }
    orf[idx] = so - si;
}

// ---------------------------------------------------------------------------
// K1/K2a: dst[b,c,l] = (relu?)( sum_c' W[c,c'] * src[b,c',l] + bias[c] )
// ---------------------------------------------------------------------------
__global__ void small_mm(const float* __restrict__ src,
                         const float* __restrict__ W,
                         const float* __restrict__ bias,
                         float* __restrict__ dst, int relu)
{
    int idx = blockIdx.x * blockDim.x + threadIdx.x;
    if (idx >= (int)SMALL) return;
    int l = idx % LL;
    int c = (idx / LL) % CC;
    int b = idx / (LL * CC);
    float acc = bias ? bias[c] : 0.f;
    const float* s = src + (size_t)b * CC * LL + l;
    const float* w = W + (size_t)c * CC;
    for (int cp = 0; cp < CC; ++cp) acc += w[cp] * s[(size_t)cp * LL];
    if (relu) acc = fmaxf(acc, 0.f);
    dst[idx] = acc;
}

// ---------------------------------------------------------------------------
// K2b: out[b,c,n,l] = adp[n] * t[b,c,l] + bias[c]
// ---------------------------------------------------------------------------
__global__ void bcast_o2r(const float* __restrict__ t,
                          const float* __restrict__ adp,
                          const float* __restrict__ bias,
                          float* __restrict__ out)
{
    size_t idx = (size_t)blockIdx.x * blockDim.x + threadIdx.x;
    if (idx >= TEN) return;
    int l = (int)(idx % LL);
    size_t r = idx / LL;
    int n = (int)(r % NN);
    size_t bc = r / NN;
    int c = (int)(bc % CC);
    out[idx] = adp[n] * t[bc * LL + l] + bias[c];
}

// ---------------------------------------------------------------------------
// K3: nconv  Y[bc,w,l] = sum_v X[bc,v,l] * Adj[v,w]   (transA: Adj[w,v])
// One block (13 waves) per bc.  X slice (200x24 = 19.2 KB) staged in LDS
// once via float4 loads; each wave owns one 16-row w-tile and streams its
// WMMA B-operands from LDS.
// A layout (16x4 f32): h=lane/16, m=lane%16; VGPR v holds K = v + 2h.
// B layout (4x16 f32): VGPR v holds row K = v + 2h, col = lane%16.
// D layout: VGPR r holds row M = r + 8h, col = lane%16.
// ---------------------------------------------------------------------------
#define NCONV_THREADS (13 * 32)

__global__ void __launch_bounds__(NCONV_THREADS)
nconv_wmma(const float* __restrict__ X,
           const float* __restrict__ A,
           float* __restrict__ Y, int transA)
{
    __shared__ float sX[NL];           // 19200 bytes

    const int bc  = blockIdx.x;        // 0..B*C-1
    const int tid = threadIdx.x;       // 0..415

    // cooperative stage of the X slice (1200 float4)
    {
        const float4* src = (const float4*)(X + (size_t)bc * NL);
        float4* dst = (float4*)sX;
        for (int i = tid; i < NL / 4; i += NCONV_THREADS) dst[i] = src[i];
    }
    __syncthreads();

    const int wt   = tid >> 5;         // wave id == w-tile, 0..12
    const int lane = tid & 31;
    const int lm   = lane & 15;
    const int h    = lane >> 4;
    const int w0   = wt * 16;

    v8f acc0 = {};
    v8f acc1 = {};
    const int  m    = w0 + lm;
    const bool mok  = (m < NN);
    const bool c1ok = (16 + lm) < LL;

    for (int k = 0; k < NN; k += 4) {
        int k0 = k + 2 * h;
        v2f av;
        if (!transA) {
            av.x = mok ? A[(size_t)k0 * NN + m]       : 0.0f;
            av.y = mok ? A[(size_t)(k0 + 1) * NN + m] : 0.0f;
        } else {
            av.x = mok ? A[(size_t)m * NN + k0]     : 0.0f;
            av.y = mok ? A[(size_t)m * NN + k0 + 1] : 0.0f;
        }
        v2f b0, b1;
        b0.x = sX[k0 * LL + lm];
        b0.y = sX[(k0 + 1) * LL + lm];
        b1.x = c1ok ? sX[k0 * LL + 16 + lm]       : 0.0f;
        b1.y = c1ok ? sX[(k0 + 1) * LL + 16 + lm] : 0.0f;

        acc0 = __builtin_amdgcn_wmma_f32_16x16x4_f32(false, av, false, b0,
                                                     (short)0, acc0, false, false);
        acc1 = __builtin_amdgcn_wmma_f32_16x16x4_f32(false, av, false, b1,
                                                     (short)0, acc1, false, false);
    }

    float* Ys = Y + (size_t)bc * NL;
    for (int r = 0; r < 8; ++r) {
        int mm = w0 + r + 8 * h;
        if (mm >= NN) continue;
        Ys[(size_t)mm * LL + lm] = acc0[r];
        if (c1ok) Ys[(size_t)mm * LL + 16 + lm] = acc1[r];
    }
}

// ---------------------------------------------------------------------------
// K4: conv1x1 partial, fused with bias / residual / per-(c,n) scale, and
// accumulated into a fuse buffer:
//   fuse[b,m,col] (+)= scale[m,n] * ( sum_c W[m, part*C+c]*X[b,c,col]
//                                      + bias[m] + residual[b,m,col] )
// One block (4 waves) per (col-tile, b); the 56x16 X tile (3.5 KB) is staged
// in LDS once, each wave owns one 16-row m-tile. K=C=56 in steps of 4.
// ---------------------------------------------------------------------------
#define CONV_THREADS (4 * 32)

__global__ void __launch_bounds__(CONV_THREADS)
convacc_wmma(const float* __restrict__ X,
             const float* __restrict__ W, int part,
             const float* __restrict__ bias,
             const float* __restrict__ residual,
             const float* __restrict__ scale,
             float* __restrict__ fuse, int init)
{
    __shared__ float sX[CC * 16];      // 3584 bytes, row-major 16 cols/row

    const int ct  = blockIdx.x;        // column tile over N*L (0..299)
    const int b   = blockIdx.z;
    const int tid = threadIdx.x;       // 0..127
    const int col0 = ct * 16;
    const int K7   = 7 * CC;

    const float* Xb = X + (size_t)b * CC * NL;

    // cooperative stage: 56 rows x 16 cols = 224 float4
    for (int i = tid; i < CC * 4; i += CONV_THREADS) {
        int row = i >> 2, q = i & 3;
        const float4* p = (const float4*)(Xb + (size_t)row * NL + col0);
        ((float4*)sX)[i] = p[q];
    }
    __syncthreads();

    const int mt   = tid >> 5;         // wave id == m-tile, 0..3
    const int lane = tid & 31;
    const int lm   = lane & 15;
    const int h    = lane >> 4;
    const int m0   = mt * 16;

    v8f acc = {};
    const int  mA  = m0 + lm;
    const bool mok = (mA < CC);

    for (int k = 0; k < CC; k += 4) {
        int k0 = k + 2 * h;
        v2f av;
        av.x = mok ? W[(size_t)mA * K7 + (size_t)part * CC + k0]     : 0.0f;
        av.y = mok ? W[(size_t)mA * K7 + (size_t)part * CC + k0 + 1] : 0.0f;
        v2f bv;
        bv.x = sX[k0 * 16 + lm];
        bv.y = sX[(k0 + 1) * 16 + lm];
        acc = __builtin_amdgcn_wmma_f32_16x16x4_f32(false, av, false, bv,
                                                    (short)0, acc, false, false);
    }

    float* Fb = fuse + (size_t)b * CC * NL;
    const float* Rb = residual ? residual + (size_t)b * CC * NL : nullptr;
    for (int r = 0; r < 8; ++r) {
        int m = m0 + r + 8 * h;
        if (m >= CC) continue;
        int col = col0 + lm;
        int n   = col / LL;
        float v = acc[r];
        if (bias) v += bias[m];
        if (Rb)   v += Rb[(size_t)m * NL + col];
        float outv = scale[(size_t)m * NN + n] * v;
        size_t idx = (size_t)m * NL + col;
        if (init) Fb[idx] = outv;
        else      Fb[idx] += outv;
    }
}

// ---------------------------------------------------------------------------
// K5: strided tconv + tanh*sigmoid gate.
// ---------------------------------------------------------------------------
__global__ void tconv_gate(const float* __restrict__ fuse,
                           const float* __restrict__ fw, const float* __restrict__ fb,
                           const float* __restrict__ gw, const float* __restrict__ gb,
                           float* __restrict__ out)
{
    const int total = BB * CC * NN * LOUT;
    int idx = blockIdx.x * blockDim.x + threadIdx.x;
    if (idx >= total) return;
    int t = idx % LOUT;
    int n = (idx / LOUT) % NN;
    int o = (idx / (LOUT * NN)) % CC;
    int b = idx / (LOUT * NN * CC);

    const float* Xp = fuse + (size_t)b * CC * NL + (size_t)n * LL + 2 * t;
    const float* fwo = fw + (size_t)o * CC * 2;
    const float* gwo = gw + (size_t)o * CC * 2;
    float f = fb[o], g = gb[o];
    for (int c = 0; c < CC; ++c) {
        float x0 = Xp[(size_t)c * NL];
        float x1 = Xp[(size_t)c * NL + 1];
        f += fwo[2 * c] * x0 + fwo[2 * c + 1] * x1;
        g += gwo[2 * c] * x0 + gwo[2 * c + 1] * x1;
    }
    out[idx] = tanhf(f) * (1.0f / (1.0f + expf(-g)));
}

// ---------------------------------------------------------------------------
// Host orchestration
// ---------------------------------------------------------------------------
extern "C" void kernel_launch(void* const* d_in, const int* in_sizes, int n_in,
                              void* d_out, int out_size, void* d_ws, size_t ws_size,
                              hipStream_t stream)
{
    (void)in_sizes; (void)n_in; (void)out_size; (void)ws_size;

    const float* inflow     = (const float*)d_in[0];
    const float* outflow    = (const float*)d_in[1];
    const float* support0   = (const float*)d_in[2];
    const float* support1   = (const float*)d_in[3];
    const float* inin_adp   = (const float*)d_in[4];
    const float* outout_adp = (const float*)d_in[5];
    const float* intout_adp = (const float*)d_in[6];
    const float* or_in_adp  = (const float*)d_in[7];
    const float* or_out_adp = (const float*)d_in[8];
    const float* fem1_ii_w = (const float*)d_in[9];
    const float* fem1_ii_b = (const float*)d_in[10];
    const float* fem1_io_w = (const float*)d_in[11];
    const float* fem1_io_b = (const float*)d_in[12];
    const float* fem1_oo_w = (const float*)d_in[13];
    const float* fem1_oo_b = (const float*)d_in[14];
    const float* fem1_oi_w = (const float*)d_in[15];
    const float* fem1_oi_b = (const float*)d_in[16];
    const float* fem2_ii_w = (const float*)d_in[17];
    const float* fem2_ii_b = (const float*)d_in[18];
    const float* fem2_io_w = (const float*)d_in[19];
    const float* fem2_io_b = (const float*)d_in[20];
    const float* fem2_oo_w = (const float*)d_in[21];
    const float* fem2_oo_b = (const float*)d_in[22];
    const float* fem2_oi_w = (const float*)d_in[23];
    const float* fem2_oi_b = (const float*)d_in[24];
    const float* split_in_w  = (const float*)d_in[25];
    const float* split_in_b  = (const float*)d_in[26];
    const float* split_out_w = (const float*)d_in[27];
    const float* split_out_b = (const float*)d_in[28];
    const float* orc_in_w  = (const float*)d_in[29];
    const float* orc_in_b  = (const float*)d_in[30];
    const float* orc_out_w = (const float*)d_in[31];
    const float* orc_out_b = (const float*)d_in[32];
    const float* w1_in  = (const float*)d_in[33];
    const float* w2_in  = (const float*)d_in[34];
    const float* w3_in  = (const float*)d_in[35];
    const float* w4_in  = (const float*)d_in[36];
    const float* w1_out = (const float*)d_in[37];
    const float* w2_out = (const float*)d_in[38];
    const float* w3_out = (const float*)d_in[39];
    const float* w4_out = (const float*)d_in[40];
    const float* filt_in_w  = (const float*)d_in[41];
    const float* filt_in_b  = (const float*)d_in[42];
    const float* gate_in_w  = (const float*)d_in[43];
    const float* gate_in_b  = (const float*)d_in[44];
    const float* filt_out_w = (const float*)d_in[45];
    const float* filt_out_b = (const float*)d_in[46];
    const float* gate_out_w = (const float*)d_in[47];
    const float* gate_out_b = (const float*)d_in[48];

    // workspace layout (floats): 5*SMALL + 6*TEN ~= 207 MB
    float* ws      = (float*)d_ws;
    float* orf     = ws;
    float* or_in   = ws + SMALL;
    float* or_out  = ws + 2 * SMALL;
    float* t_in    = ws + 3 * SMALL;
    float* t_out   = ws + 4 * SMALL;
    float* o2r_in  = ws + 5 * SMALL;
    float* o2r_out = o2r_in  + TEN;
    float* tmp1    = o2r_out + TEN;
    float* tmp2    = tmp1    + TEN;
    float* fuse_in = tmp2    + TEN;
    float* fuse_out= fuse_in + TEN;

    const dim3 blk(256);
    const int gridSmall = (int)((SMALL + 255) / 256);
    const unsigned gridTen = (unsigned)((TEN + 255) / 256);

    // orf + origin/return split path
    orf_kernel<<<gridSmall, blk, 0, stream>>>(inflow, outflow, orf);
    small_mm<<<gridSmall, blk, 0, stream>>>(orf, split_in_w,  split_in_b,  or_in,  1);
    small_mm<<<gridSmall, blk, 0, stream>>>(orf, split_out_w, split_out_b, or_out, 1);
    small_mm<<<gridSmall, blk, 0, stream>>>(or_in,  orc_in_w,  nullptr, t_in,  0);
    small_mm<<<gridSmall, blk, 0, stream>>>(or_out, orc_out_w, nullptr, t_out, 0);
    bcast_o2r<<<gridTen, blk, 0, stream>>>(t_in,  or_in_adp,  orc_in_b,  o2r_in);
    bcast_o2r<<<gridTen, blk, 0, stream>>>(t_out, or_out_adp, orc_out_b, o2r_out);

    const dim3 gN(BB * CC);                 // nconv: one block (13 waves) per bc
    const dim3 bN(NCONV_THREADS);
    const dim3 g4(NL / 16, 1, BB);          // convacc: 300 col-tiles x B
    const dim3 b4(CONV_THREADS);

    auto gcn = [&](const float* x, const float* a2, int tr2,
                   const float* Wt, const float* bias,
                   const float* scale, float* fuse, int init) {
        // part 0: identity part, carries bias + residual(x)
        convacc_wmma<<<g4, b4, 0, stream>>>(x, Wt, 0, bias, x, scale, fuse, init);
        const float* sup[3] = {support0, support1, a2};
        const int    trs[3] = {0, 0, tr2};
        for (int s = 0; s < 3; ++s) {
            nconv_wmma<<<gN, bN, 0, stream>>>(x, sup[s], tmp1, trs[s]);
            convacc_wmma<<<g4, b4, 0, stream>>>(tmp1, Wt, 2 * s + 1,
                                                nullptr, nullptr, scale, fuse, 0);
            nconv_wmma<<<gN, bN, 0, stream>>>(tmp1, sup[s], tmp2, trs[s]);
            convacc_wmma<<<g4, b4, 0, stream>>>(tmp2, Wt, 2 * s + 2,
                                                nullptr, nullptr, scale, fuse, 0);
        }
    };

    // fem1 (inputs: inflow / outflow)
    gcn(inflow,  inin_adp,   0, fem1_ii_w, fem1_ii_b, w1_in,  fuse_in,  1);
    gcn(inflow,  intout_adp, 0, fem1_io_w, fem1_io_b, w2_in,  fuse_in,  0);
    gcn(outflow, outout_adp, 0, fem1_oo_w, fem1_oo_b, w1_out, fuse_out, 1);
    gcn(outflow, intout_adp, 1, fem1_oi_w, fem1_oi_b, w2_out, fuse_out, 0);
    // fem2 (inputs: o2r_in / o2r_out)
    gcn(o2r_in,  inin_adp,   0, fem2_ii_w, fem2_ii_b, w3_in,  fuse_in,  0);
    gcn(o2r_in,  intout_adp, 0, fem2_io_w, fem2_io_b, w4_out, fuse_out, 0);
    gcn(o2r_out, outout_adp, 0, fem2_oo_w, fem2_oo_b, w3_out, fuse_out, 0);
    gcn(o2r_out, intout_adp, 1, fem2_oi_w, fem2_oi_b, w4_in,  fuse_in,  0);

    // temporal conv + gating -> outputs (new_in, new_out) concatenated
    float* out = (float*)d_out;
    const size_t halfOut = (size_t)BB * CC * NN * LOUT;
    const int gridOut = (int)((halfOut + 255) / 256);
    tconv_gate<<<gridOut, blk, 0, stream>>>(fuse_in,  filt_in_w,  filt_in_b,
                                            gate_in_w,  gate_in_b,  out);
    tconv_gate<<<gridOut, blk, 0, stream>>>(fuse_out, filt_out_w, filt_out_b,
                                            gate_out_w, gate_out_b, out + halfOut);
}